// StepWiseTrainablePulseShaping_30889404792872
// MI455X (gfx1250) — compile-verified
//
#include <hip/hip_runtime.h>
#include <hip/hip_bf16.h>

// ---- problem constants (static in the reference) ----
#define M_TAPS   28673
#define D_LAG    4096
#define NLAGS    (2 * D_LAG - 1)      // 8191 lags
#define SSHIFT   (7 * (D_LAG - 1))    // 28665
#define L_OUT    16383                // output complex length
#define KSPLIT   8                    // k-slices (waves) per lag group
#define N0_START (-1680)              // -112*15 so jb=15 column covers n=0
// sweep n0 = N0_START + 4*t ; need max n0 >= M-4 = 28669 -> t_max = 7588.
// Round total iterations up to a multiple of 2*KSPLIT (tail reads zero pad).
#define ITERS_TOTAL 7600
#define ITERS_PER   (ITERS_TOTAL / KSPLIT)   // 950 wmma ops per wave

// ---- padded staging buffers in d_ws (float offsets) ----
// wt_pad covers ib in [-1680, 30399]; wr_pad covers ia in [-28672, 57384]
#define PADT      1680
#define WT_SIZE   32080
#define PADR      28672
#define WR_SIZE   86064
#define WS_WT_OFF 16                       // ws[0] = scale; 16-float header
#define WS_WR_OFF (WS_WT_OFF + WT_SIZE)
// total ws bytes needed: (16 + 32080 + 86064) * 4 = 472,640

typedef __attribute__((ext_vector_type(2))) float v2f;
typedef __attribute__((ext_vector_type(8))) float v8f;

// Kernel 1: normalization scalar.  scale = c / (sqrt(c*St) * sqrt(c*Sr))
__global__ void norm_scale_kernel(const float* __restrict__ wtx,
                                  const float* __restrict__ wrx,
                                  float* __restrict__ ws) {
  __shared__ float s1[256];
  __shared__ float s2[256];
  const int tid = threadIdx.x;
  float st = 0.f, sr = 0.f;
  for (int i = tid; i < M_TAPS; i += 256) {
    float a = wtx[i]; st += a * a;
    float b = wrx[i]; sr += b * b;
  }
  s1[tid] = st; s2[tid] = sr;
  __syncthreads();
  for (int off = 128; off > 0; off >>= 1) {
    if (tid < off) { s1[tid] += s1[tid + off]; s2[tid] += s2[tid + off]; }
    __syncthreads();
  }
  if (tid == 0) {
    const float c = 1.0f / (float)M_TAPS;
    ws[0] = c / (sqrtf(c * s1[0]) * sqrtf(c * s2[0]));
  }
}

// Kernel 2: build zero-padded copies of both vectors in scratch, so the hot
// WMMA loop needs no bounds checks (loads become unconditional).
__global__ void pad_kernel(const float* __restrict__ wtx,
                           const float* __restrict__ wrx,
                           float* __restrict__ ws) {
  const int i = blockIdx.x * blockDim.x + threadIdx.x;
  if (i < WT_SIZE) {
    const int src = i - PADT;
    ws[WS_WT_OFF + i] = ((unsigned)src < (unsigned)M_TAPS) ? wtx[src] : 0.f;
  }
  if (i < WR_SIZE) {
    const int src = i - PADR;
    ws[WS_WR_OFF + i] = ((unsigned)src < (unsigned)M_TAPS) ? wrx[src] : 0.f;
  }
}

// Kernel 3: clear the complex64 output (re+im interleaved floats).
__global__ void zero_out_kernel(float* __restrict__ out, int n) {
  int i = blockIdx.x * blockDim.x + threadIdx.x;
  if (i < n) out[i] = 0.f;
}

// Kernel 4: strided correlation as WMMA GEMM.
// Block g handles lags j = 256*g + 16*jb + ji. Each of the 8 waves sweeps a
// contiguous slice of n0 with TWO independent accumulator chains (even/odd
// chunks) to break the D->C serial dependency; partial tiles combined in LDS.
__global__ __launch_bounds__(256) void corr_wmma_kernel(
    const float* __restrict__ ws, const int* __restrict__ Lp,
    float* __restrict__ out) {
  __shared__ float red[KSPLIT * 256];

  const float* __restrict__ wtp = ws + WS_WT_OFF + PADT;  // wt_pad, index by ib
  const float* __restrict__ wrp = ws + WS_WR_OFF + PADR;  // wr_pad, index by ia

  const int g    = blockIdx.x;       // lag group (32 groups of 256 lags)
  const int tid  = threadIdx.x;
  const int wave = tid >> 5;         // wave32
  const int lane = tid & 31;
  const int half = lane >> 4;
  const int r    = lane & 15;        // A-row ji for A, B-col jb for B
  const int kk   = half * 2;         // this half-wave holds K = kk, kk+1

  // A[ji,k] = wr[n0 + k + SSHIFT - 1792*g - 7*ji]
  // B[k,jb] = wt[n0 + k + 112*jb]
  const int n0_0 = N0_START + 4 * (wave * ITERS_PER);
  int ia = n0_0 + kk + SSHIFT - 1792 * g - 7 * r;
  int ib = n0_0 + kk + 112 * r;

  v8f acc0 = {};
  v8f acc1 = {};
  for (int it = 0; it < ITERS_PER / 2; ++it) {
    // chain 0: chunk n0
    float a0 = wrp[ia];     float a1 = wrp[ia + 1];
    float b0 = wtp[ib];     float b1 = wtp[ib + 1];
    v2f av0 = {a0, a1};
    v2f bv0 = {b0, b1};
    acc0 = __builtin_amdgcn_wmma_f32_16x16x4_f32(
        false, av0, false, bv0, (short)0, acc0, false, false);
    // chain 1: chunk n0+4
    float a2 = wrp[ia + 4]; float a3 = wrp[ia + 5];
    float b2 = wtp[ib + 4]; float b3 = wtp[ib + 5];
    v2f av1 = {a2, a3};
    v2f bv1 = {b2, b3};
    acc1 = __builtin_amdgcn_wmma_f32_16x16x4_f32(
        false, av1, false, bv1, (short)0, acc1, false, false);
    ia += 8;
    ib += 8;
  }

  // C/D layout: element v of lane -> (ji = v + 8*half, jb = lane & 15)
  #pragma unroll
  for (int v = 0; v < 8; ++v) {
    const int ji = v + 8 * half;
    const int jb = r;
    red[wave * 256 + jb * 16 + ji] = acc0[v] + acc1[v];
  }
  __syncthreads();

  const float scale = ws[0];
  const int   L     = Lp[0];
  const int   pad   = (L - 2 * D_LAG + 1) >> 1;   // 4096 for L=16383

  // deterministic cross-slice reduction: one lag per thread
  float sum = 0.f;
  #pragma unroll
  for (int w = 0; w < KSPLIT; ++w) sum += red[w * 256 + tid];

  const int j = g * 256 + tid;                    // lag index in [0, 8192)
  if (j < NLAGS) out[2 * (pad + j)] = scale * sum; // real part; imag stays 0
}

extern "C" void kernel_launch(void* const* d_in, const int* in_sizes, int n_in,
                              void* d_out, int out_size, void* d_ws, size_t ws_size,
                              hipStream_t stream) {
  (void)in_sizes; (void)n_in; (void)out_size; (void)ws_size;
  const float* wtx = (const float*)d_in[0];
  const float* wrx = (const float*)d_in[1];
  const int*   Lp  = (const int*)d_in[2];
  float* out = (float*)d_out;
  float* ws  = (float*)d_ws;

  norm_scale_kernel<<<1, 256, 0, stream>>>(wtx, wrx, ws);

  const int padN = (WR_SIZE + 255) / 256;   // WR_SIZE > WT_SIZE
  pad_kernel<<<padN, 256, 0, stream>>>(wtx, wrx, ws);

  const int totalFloats = 2 * L_OUT;  // complex64 -> interleaved floats
  zero_out_kernel<<<(totalFloats + 255) / 256, 256, 0, stream>>>(out, totalFloats);

  corr_wmma_kernel<<<32, 256, 0, stream>>>(ws, Lp, out);
}